// ContrastiveLoss_9990093931268
// MI455X (gfx1250) — compile-verified
//
#include <hip/hip_runtime.h>
#include <hip/hip_bf16.h>

typedef __attribute__((ext_vector_type(2))) float v2f;
typedef __attribute__((ext_vector_type(4))) float v4f;
typedef __attribute__((ext_vector_type(8))) float v8f;

#define N_PTS 8192
#define DIM   128
#define EPS_C 1.0f
// N*(N-1)/2 = 8192*8191/2
#define INV_DEN (1.0f / 33550336.0f)

// ---------------------------------------------------------------------------
// Zero the scalar output (harness poisons d_out before timing).
// ---------------------------------------------------------------------------
__global__ void cl_init_out(float* out) {
    if (threadIdx.x == 0 && blockIdx.x == 0) out[0] = 0.0f;
}

// ---------------------------------------------------------------------------
// sq[i] = sum_k ys[i][k]^2.  One wave32 per row; lane l sums 4 floats.
// ---------------------------------------------------------------------------
__global__ void cl_sqnorm(const float* __restrict__ ys, float* __restrict__ sq) {
    const int row  = blockIdx.x * 8 + (threadIdx.x >> 5);  // 8 waves per block
    const int lane = threadIdx.x & 31;
    const v4f* p = reinterpret_cast<const v4f*>(ys + (size_t)row * DIM);
    v4f v = p[lane];                                       // 32 lanes * 4 = 128
    float s = v.x * v.x + v.y * v.y + v.z * v.z + v.w * v.w;
#pragma unroll
    for (int off = 16; off > 0; off >>= 1) s += __shfl_xor(s, off, 32);
    if (lane == 0) sq[row] = s;
}

// ---------------------------------------------------------------------------
// Main kernel: wave-per-(64x64) tile-pair over the upper triangle.
//   G = Y * Y^T accumulated with V_WMMA_F32_16X16X4_F32 (fp32 matrix cores),
//   4x4 register blocking: 8 fragment loads feed 16 WMMAs per K-step.
//   Epilogue: d2 = sq_i + sq_j - 2*dot; same-label -> d2, else hinge^2.
// Grid: (T/2, T/4) blocks of 256 threads (8 waves); wave w covers
//   ti = bx*2 + (w&1), tj = by*4 + (w>>1); waves with tj<ti exit (uniform).
// ---------------------------------------------------------------------------
__global__ void __launch_bounds__(256)
cl_pairs_wmma(const float* __restrict__ ys,
              const int*   __restrict__ labels,
              const float* __restrict__ sq,
              float*       __restrict__ out) {
    const int wave = threadIdx.x >> 5;
    const int lane = threadIdx.x & 31;

    const int ti = blockIdx.x * 2 + (wave & 1);   // 64-row tile index (i side)
    const int tj = blockIdx.y * 4 + (wave >> 1);  // 64-row tile index (j side)
    if (tj < ti) return;                          // wave-uniform: EXEC stays full

    // Fragment addressing per ISA 7.12.2 (32-bit A 16x4):
    //   lanes 0-15 hold M=0..15 with K={k0,k0+1}; lanes 16-31 hold K={k0+2,k0+3}.
    const int lm    = lane & 15;        // M (A) or N (B) within 16x16
    const int khalf = lane >> 4;        // selects K pair within the 4-wide step

    const float* __restrict__ arow[4];
    const float* __restrict__ brow[4];
#pragma unroll
    for (int t = 0; t < 4; ++t) {
        arow[t] = ys + (size_t)(ti * 64 + t * 16 + lm) * DIM + 2 * khalf;
        brow[t] = ys + (size_t)(tj * 64 + t * 16 + lm) * DIM + 2 * khalf;
    }

    v8f acc[4][4];
#pragma unroll
    for (int mi = 0; mi < 4; ++mi)
#pragma unroll
        for (int nj = 0; nj < 4; ++nj)
            acc[mi][nj] = (v8f){};

#pragma unroll 2
    for (int ks = 0; ks < DIM / 4; ++ks) {
        const int k = 4 * ks;
        v2f a[4], b[4];
#pragma unroll
        for (int t = 0; t < 4; ++t) {
            a[t] = *reinterpret_cast<const v2f*>(arow[t] + k);
            b[t] = *reinterpret_cast<const v2f*>(brow[t] + k);
        }
#pragma unroll
        for (int mi = 0; mi < 4; ++mi)
#pragma unroll
            for (int nj = 0; nj < 4; ++nj)
                acc[mi][nj] = __builtin_amdgcn_wmma_f32_16x16x4_f32(
                    false, a[mi], false, b[nj], (short)0, acc[mi][nj], false, false);
    }

    // Epilogue. C/D layout: VGPR v, lanes 0-15 -> M=v, lanes 16-31 -> M=v+8; N=lane&15.
    float local = 0.0f;

#pragma unroll
    for (int nj = 0; nj < 4; ++nj) {
        const int   gj = tj * 64 + nj * 16 + lm;
        const int   lj = labels[gj];
        const float sj = sq[gj];
#pragma unroll
        for (int mi = 0; mi < 4; ++mi) {
#pragma unroll
            for (int v = 0; v < 8; ++v) {
                const int gi = ti * 64 + mi * 16 + v + 8 * khalf;
                if (gi < gj) {
                    const float si = sq[gi];
                    float d2 = si + sj - 2.0f * acc[mi][nj][v];
                    d2 = fmaxf(d2, 0.0f);
                    if (labels[gi] == lj) {
                        local += d2;                      // POS_WEIGHT = 1
                    } else {
                        const float t = EPS_C - __builtin_sqrtf(d2);
                        if (t > 0.0f) local += t * t;
                    }
                }
            }
        }
    }

#pragma unroll
    for (int off = 16; off > 0; off >>= 1) local += __shfl_xor(local, off, 32);
    if (lane == 0) atomicAdd(out, local * INV_DEN);
}

// ---------------------------------------------------------------------------
extern "C" void kernel_launch(void* const* d_in, const int* in_sizes, int n_in,
                              void* d_out, int out_size, void* d_ws, size_t ws_size,
                              hipStream_t stream) {
    const float* ys     = (const float*)d_in[0];
    const int*   labels = (const int*)d_in[1];
    float*       out    = (float*)d_out;
    float*       sq     = (float*)d_ws;  // 8192 floats = 32 KB scratch

    cl_init_out<<<1, 64, 0, stream>>>(out);

    // 8192 rows / 8 waves-per-block = 1024 blocks
    cl_sqnorm<<<N_PTS / 8, 256, 0, stream>>>(ys, sq);

    // T = 8192/64 = 128 tiles per dim; waves cover 2x4 tiles per block
    dim3 grid(128 / 2, 128 / 4);
    cl_pairs_wmma<<<grid, 256, 0, stream>>>(ys, labels, sq, out);
}